// FeaturePropagator_44384192037433
// MI455X (gfx1250) — compile-verified
//
#include <hip/hip_runtime.h>
#include <stdint.h>

#define N_NODES_C 100000
#define N_EDGES_C 1600000
#define D_FEAT_C  64
#define NUM_ITERS_C 10

// ---------------------------------------------------------------------------
// CDNA5 inline-asm helpers
// ---------------------------------------------------------------------------

// Per-lane async gather of 16 bytes from global memory directly into LDS.
// LDS byte address comes from each lane's VGPR; tracked by ASYNCcnt.
__device__ __forceinline__ void async_gather_b128(uint32_t lds_byte_addr,
                                                  const float* gsrc) {
    asm volatile("global_load_async_to_lds_b128 %0, %1, off"
                 :
                 : "v"(lds_byte_addr), "v"(gsrc)
                 : "memory");
}

__device__ __forceinline__ void wait_async_le1() {
    asm volatile("s_wait_asynccnt 0x1" ::: "memory");
}
__device__ __forceinline__ void wait_async_le0() {
    asm volatile("s_wait_asynccnt 0x0" ::: "memory");
}

// ---------------------------------------------------------------------------
// One-time preprocessing kernels
// ---------------------------------------------------------------------------

__global__ void k_zero4(float4* __restrict__ p, int n4) {
    int i = blockIdx.x * blockDim.x + threadIdx.x;
    if (i < n4) p[i] = make_float4(0.f, 0.f, 0.f, 0.f);
}

// histogram of idx[] into cnt[] (used for col-degree and row-CSR counts)
__global__ void k_count(const int* __restrict__ idx,
                        unsigned int* __restrict__ cnt, int n) {
    int i = blockIdx.x * blockDim.x + threadIdx.x;
    if (i < n) atomicAdd(&cnt[idx[i]], 1u);
}

// In-place: reinterpret integer degree -> deg_inv_sqrt (0 when deg == 0)
__global__ void k_deg_inv_sqrt(float* __restrict__ degdis, int n) {
    int i = blockIdx.x * blockDim.x + threadIdx.x;
    if (i < n) {
        unsigned int c = reinterpret_cast<unsigned int*>(degdis)[i];
        degdis[i] = (c > 0u) ? rsqrtf((float)c) : 0.0f;
    }
}

// Single-block exclusive scan: ptr[i] = sum(cnt[0..i)), ptr[n] = total.
// Also writes the exclusive value to ofs[] (may alias cnt: read-then-write).
__global__ __launch_bounds__(1024)
void k_scan_ptr(const unsigned int* __restrict__ cnt, int* __restrict__ ptr,
                unsigned int* __restrict__ ofs, int n) {
    __shared__ unsigned int s[1024];
    __shared__ unsigned int sbase;
    const int tid = threadIdx.x;
    if (tid == 0) sbase = 0u;
    __syncthreads();
    for (int base = 0; base < n; base += 1024) {
        const int i = base + tid;
        const unsigned int v = (i < n) ? cnt[i] : 0u;
        s[tid] = v;
        __syncthreads();
        for (int off = 1; off < 1024; off <<= 1) {
            unsigned int t = (tid >= off) ? s[tid - off] : 0u;
            __syncthreads();
            s[tid] += t;
            __syncthreads();
        }
        const unsigned int incl = s[tid];
        const unsigned int excl = incl - v;
        const unsigned int b = sbase;
        if (i < n) {
            ptr[i] = (int)(b + excl);
            ofs[i] = b + excl;
        }
        __syncthreads();
        if (tid == 1023) sbase = b + incl;
        __syncthreads();
    }
    if (tid == 0) ptr[n] = (int)sbase;
}

// Scatter edges into CSR order; precompute w = dis[row]*dis[col] per edge.
__global__ void k_fill_csr(const int* __restrict__ row, const int* __restrict__ col,
                           const float* __restrict__ dis, unsigned int* __restrict__ ofs,
                           int* __restrict__ csr_col, float* __restrict__ csr_w, int n) {
    int e = blockIdx.x * blockDim.x + threadIdx.x;
    if (e < n) {
        const int r = row[e];
        const int c = col[e];
        const unsigned int pos = atomicAdd(&ofs[r], 1u);
        csr_col[pos] = c;
        csr_w[pos]   = dis[r] * dis[c];
    }
}

// cur0 = mask ? x : 0   (float4 granularity; 16 chunks per node)
__global__ void k_init_masked(const float4* __restrict__ x,
                              const int* __restrict__ mask,
                              float4* __restrict__ out, int n4) {
    int i = blockIdx.x * blockDim.x + threadIdx.x;
    if (i < n4) {
        int node = i >> 4;
        out[i] = mask[node] ? x[i] : make_float4(0.f, 0.f, 0.f, 0.f);
    }
}

// ---------------------------------------------------------------------------
// CSR gather-reduce SpMM (atomic-free hot loop), masked clamp fused.
//
// One wave per node. 16 lanes x float4 cover the 64-float feature row;
// the two half-waves process 2 edges per step. Feature rows of the NEXT
// edge pair are gathered into per-wave-private LDS with
// GLOBAL_LOAD_ASYNC_TO_LDS_B128 (double buffered, s_wait_asynccnt) while
// the current pair accumulates into registers. Halves combined with
// shfl_xor(16); final store applies out[i] = mask[i] ? x[i] : acc.
// ---------------------------------------------------------------------------
__global__ __launch_bounds__(256)
void k_spmm_csr(const float* __restrict__ cur, float* __restrict__ out,
                const int* __restrict__ ptr, const int* __restrict__ csr_col,
                const float* __restrict__ csr_w, const float* __restrict__ x,
                const int* __restrict__ mask) {
    // [wave(8)][stage(2)][edge(2)][feat(64)] floats = 8 KB
    __shared__ __align__(16) float sh[8 * 2 * 2 * 64];

    const int tid   = threadIdx.x;
    const int wave  = tid >> 5;
    const int lane  = tid & 31;
    const int halfe = lane >> 4;   // which edge of the current pair
    const int sub   = lane & 15;   // float4 chunk within the feature row

    float* shWave = &sh[wave * 256];
    const uint32_t ldsLane =
        (uint32_t)(uintptr_t)shWave + (uint32_t)(halfe * 64 + sub * 4) * 4u;
    // stage stride = 128 floats = 512 bytes

    const int waveGlobal = blockIdx.x * 8 + wave;
    const int nWaves     = gridDim.x * 8;

    for (int i = waveGlobal; i < N_NODES_C; i += nWaves) {
        const int start = ptr[i];
        const int end   = ptr[i + 1];
        float4 acc = make_float4(0.f, 0.f, 0.f, 0.f);

        const int nst = (end - start + 1) >> 1;   // edge pairs (steps)
        if (nst > 0) {
            // ---- prologue: gather first pair into stage 0 ----
            int  j     = start + halfe;
            bool v_cur = (j < end);               // halfe==0 lanes always valid
            float w_cur = 0.f;
            if (v_cur) {
                const int c = csr_col[j];
                w_cur = csr_w[j];
                async_gather_b128(ldsLane, cur + (size_t)c * D_FEAT_C + sub * 4);
            }
            int stage = 0;
            for (int s = 0; s < nst; ++s) {
                const bool more = (s + 1 < nst);
                bool  v_nxt = false;
                float w_nxt = 0.f;
                if (more) {
                    const int jn = start + (s + 1) * 2 + halfe;
                    v_nxt = (jn < end);           // halfe==0 lane always valid
                    if (v_nxt) {
                        const int c = csr_col[jn];
                        w_nxt = csr_w[jn];
                        __builtin_prefetch(&csr_col[jn + 2], 0, 0);
                        __builtin_prefetch(&csr_w[jn + 2], 0, 0);
                        async_gather_b128(ldsLane + (uint32_t)(stage ^ 1) * 512u,
                                          cur + (size_t)c * D_FEAT_C + sub * 4);
                    }
                    wait_async_le1();             // current pair's gather done
                } else {
                    wait_async_le0();
                }
                if (v_cur) {
                    const float4 v = *reinterpret_cast<const float4*>(
                        &shWave[stage * 128 + halfe * 64 + sub * 4]);
                    acc.x += w_cur * v.x;
                    acc.y += w_cur * v.y;
                    acc.z += w_cur * v.z;
                    acc.w += w_cur * v.w;
                }
                v_cur = v_nxt; w_cur = w_nxt; stage ^= 1;
            }
        }

        // combine the two half-wave partial sums
        acc.x += __shfl_xor(acc.x, 16, 32);
        acc.y += __shfl_xor(acc.y, 16, 32);
        acc.z += __shfl_xor(acc.z, 16, 32);
        acc.w += __shfl_xor(acc.w, 16, 32);

        if (halfe == 0) {
            const size_t o = (size_t)i * D_FEAT_C + sub * 4;
            float4 r;
            if (mask[i]) r = *reinterpret_cast<const float4*>(x + o);
            else         r = acc;
            *reinterpret_cast<float4*>(out + o) = r;
        }
    }
}

// ---------------------------------------------------------------------------
// Host-side launch
// ---------------------------------------------------------------------------
extern "C" void kernel_launch(void* const* d_in, const int* in_sizes, int n_in,
                              void* d_out, int out_size, void* d_ws, size_t ws_size,
                              hipStream_t stream) {
    (void)in_sizes; (void)n_in; (void)out_size; (void)ws_size;

    const float* x    = (const float*)d_in[0];          // [N, 64]
    const int*   ei   = (const int*)d_in[1];            // [2, E] flat
    const int*   mask = (const int*)d_in[2];            // [N]
    const int*   row  = ei;                             // ei[0, :]
    const int*   col  = ei + N_EDGES_C;                 // ei[1, :]

    // ws layout (float-sized units), node arrays padded to 100352 (/256)
    const int NP = 100352;
    float*        dis     = (float*)d_ws;               // N floats (deg->dis)
    unsigned int* cnt     = (unsigned int*)(dis + NP);  // N uints (then ofs)
    int*          ptr     = (int*)(dis + 2 * NP);       // N+1 ints
    int*          csr_col = (int*)(dis + 3 * NP);       // E ints
    float*        csr_w   = (float*)(dis + 3 * NP) + N_EDGES_C;  // E floats
    float*        bufA    = csr_w + N_EDGES_C;          // N*64 floats

    const int n4feat = N_NODES_C * (D_FEAT_C / 4);      // 1,600,000 float4
    dim3 blk(256);
    dim3 gN((N_NODES_C + 255) / 256);
    dim3 gN4d((N_NODES_C / 4 + 255) / 256);             // N words as float4
    dim3 gE((N_EDGES_C + 255) / 256);
    dim3 gF((n4feat + 255) / 256);

    // --- one-time preprocessing ---
    k_zero4<<<gN4d, blk, 0, stream>>>((float4*)dis, N_NODES_C / 4);
    k_zero4<<<gN4d, blk, 0, stream>>>((float4*)cnt, N_NODES_C / 4);
    k_count<<<gE, blk, 0, stream>>>(col, (unsigned int*)dis, N_EDGES_C); // deg (col)
    k_count<<<gE, blk, 0, stream>>>(row, cnt, N_EDGES_C);                // CSR counts
    k_deg_inv_sqrt<<<gN, blk, 0, stream>>>(dis, N_NODES_C);
    k_scan_ptr<<<1, 1024, 0, stream>>>(cnt, ptr, cnt /* ofs in place */, N_NODES_C);
    k_fill_csr<<<gE, blk, 0, stream>>>(row, col, dis, cnt, csr_col, csr_w, N_EDGES_C);

    // cur0 = mask ? x : 0, placed in d_out (even #iters -> result ends in d_out)
    float* cur = (float*)d_out;
    float* nxt = bufA;
    k_init_masked<<<gF, blk, 0, stream>>>((const float4*)x, mask,
                                          (float4*)cur, n4feat);

    // --- 10 atomic-free, async-pipelined SpMM iterations (clamp fused) ---
    for (int it = 0; it < NUM_ITERS_C; ++it) {
        k_spmm_csr<<<2048, blk, 0, stream>>>(cur, nxt, ptr, csr_col, csr_w,
                                             x, mask);
        float* tmp = cur; cur = nxt; nxt = tmp;
    }
    // after an even number of swaps, cur == d_out holds the final result
}